// ProjectLayerInference_81466939670809
// MI455X (gfx1250) — compile-verified
//
#include <hip/hip_runtime.h>
#include <hip/hip_bf16.h>

// Problem constants (match reference setup_inputs)
#define B_  4
#define V_  5
#define J_  15
#define H_  128
#define W_  240
#define N_  128000          // 80*80*20

#define NTHREADS 1024       // 32 wave32s per workgroup
#define PT       25         // points per thread held in register accumulators
#define NC       (NTHREADS * PT)   // 25600 points per chunk
#define CH       (N_ / NC)         // 5 chunks -> 300 workgroups total

#define SLICE    (H_ * W_)         // 30720 floats = 122,880 B per heatmap slice
#define NV4      (SLICE / 4)       // 7680 b128 transfers per slice
#define KOPS     8                 // uniform async ops per thread per slice (8*1024 >= 7680)

// ---- CDNA5 async global->LDS path (probe-verified on this toolchain) ----
#ifndef HAVE_ASYNC_B128
#  if defined(__has_builtin)
#    if __has_builtin(__builtin_amdgcn_global_load_async_to_lds_b128)
#      define HAVE_ASYNC_B128 1
#    else
#      define HAVE_ASYNC_B128 0
#    endif
#  else
#    define HAVE_ASYNC_B128 0
#  endif
#endif

#if HAVE_ASYNC_B128
#  if defined(__has_builtin) && __has_builtin(__builtin_amdgcn_s_wait_asynccnt)
#    define WAIT_ASYNC(n) __builtin_amdgcn_s_wait_asynccnt(n)
#  else
#    define WAIT_ASYNC(n) asm volatile("s_wait_asynccnt %0" ::"i"(n) : "memory")
#  endif
#else
#  define WAIT_ASYNC(n) ((void)0)
#endif

// Builtin signature (from hipcc diagnostic): (v4i addrspace(1)*, v4i addrspace(3)*, imm, imm)
typedef int v4i __attribute__((ext_vector_type(4)));
typedef __attribute__((address_space(1))) v4i* g_v4i_p;   // global
typedef __attribute__((address_space(3))) v4i* l_v4i_p;   // LDS

// Issue one full slice copy (global -> LDS). Exactly KOPS async ops per thread
// (tail index clamped) so every wave's ASYNCcnt advances by the same amount --
// required for the s_wait_asynccnt threshold accounting in the pipeline.
__device__ __forceinline__ void stage_slice(const float* __restrict__ src,
                                            float* dst, int tid)
{
#if HAVE_ASYNC_B128
#pragma unroll
    for (int k = 0; k < KOPS; ++k) {
        int i = tid + k * NTHREADS;
        i = (i < NV4) ? i : (NV4 - 1);   // duplicate-write pad, keeps count uniform
        __builtin_amdgcn_global_load_async_to_lds_b128(
            (g_v4i_p)(src + 4 * i), (l_v4i_p)(dst + 4 * i), 0, 0);
    }
#else
    for (int i = tid; i < NV4; i += NTHREADS) {
        float4 t = reinterpret_cast<const float4*>(src)[i];
        reinterpret_cast<float4*>(dst)[i] = t;
    }
#endif
}

__global__ __launch_bounds__(NTHREADS)
void project_sample_kernel(const float* __restrict__ hm,     // (B,V,J,H,W)
                           const float* __restrict__ grid,   // (B,V,1,N,2)
                           float* __restrict__ out)          // (B,J,N)
{
    __shared__ float smem[2][SLICE];   // 245,760 B double buffer (< 320 KB/WGP)

    const int tid = threadIdx.x;
    const int bid = blockIdx.x;
    const int c = bid % CH;
    const int j = (bid / CH) % J_;
    const int b = bid / (CH * J_);
    const int base_pt = c * NC;

    const float half_wm1 = 0.5f * (float)(W_ - 1);
    const float half_hm1 = 0.5f * (float)(H_ - 1);
    const float wmax = (float)(W_ - 1);
    const float hmax = (float)(H_ - 1);

    const float* slice0 = hm + (((size_t)(b * V_ + 0) * J_ + j) * (size_t)SLICE);

    float acc[PT];
#pragma unroll
    for (int p = 0; p < PT; ++p) acc[p] = 0.0f;

    // ---- prologue: start streaming view 0 into buffer 0 ----
    stage_slice(slice0, smem[0], tid);

    for (int v = 0; v < V_; ++v) {
        const int cur = v & 1;

        // ---- issue next view's slice into the other buffer (overlapped) ----
        // Safe: that buffer was last *read* in iteration v-1, and all waves
        // passed the end-of-iteration barrier before entering iteration v.
        if (v + 1 < V_) {
            const float* nxt = hm + (((size_t)(b * V_ + (v + 1)) * J_ + j) * (size_t)SLICE);
            stage_slice(nxt, smem[1 - cur], tid);
            // Wait until only the KOPS just-issued ops remain outstanding.
            // Async loads complete in order, so the current buffer has landed.
            WAIT_ASYNC(KOPS);
        } else {
            WAIT_ASYNC(0);
        }
        __syncthreads();   // every wave's async writes to smem[cur] are visible

        // ---- bilinear gather of this chunk's points from LDS ----
        const float2* g2 =
            reinterpret_cast<const float2*>(grid + ((size_t)(b * V_ + v) * N_ + base_pt) * 2);
        const float* buf = smem[cur];
#pragma unroll
        for (int p = 0; p < PT; ++p) {
            const int idx = p * NTHREADS + tid;
            const float2 g = g2[idx];

            const float ix = (g.x + 1.0f) * half_wm1;
            const float iy = (g.y + 1.0f) * half_hm1;
            const float x0f = floorf(ix);
            const float y0f = floorf(iy);
            const float x1f = x0f + 1.0f;
            const float y1f = y0f + 1.0f;
            const float wx1 = ix - x0f, wx0 = 1.0f - wx1;
            const float wy1 = iy - y0f, wy0 = 1.0f - wy1;

            // zero-padding validity masks (exactly mirrors reference)
            const float mx0 = (x0f >= 0.0f && x0f <= wmax) ? 1.0f : 0.0f;
            const float mx1 = (x1f >= 0.0f && x1f <= wmax) ? 1.0f : 0.0f;
            const float my0 = (y0f >= 0.0f && y0f <= hmax) ? 1.0f : 0.0f;
            const float my1 = (y1f >= 0.0f && y1f <= hmax) ? 1.0f : 0.0f;

            const int xi0 = (int)fminf(fmaxf(x0f, 0.0f), wmax);
            const int xi1 = (int)fminf(fmaxf(x1f, 0.0f), wmax);
            const int yi0 = (int)fminf(fmaxf(y0f, 0.0f), hmax);
            const int yi1 = (int)fminf(fmaxf(y1f, 0.0f), hmax);

            const float v00 = buf[yi0 * W_ + xi0];
            const float v01 = buf[yi0 * W_ + xi1];
            const float v10 = buf[yi1 * W_ + xi0];
            const float v11 = buf[yi1 * W_ + xi1];

            float s = v00 * (wx0 * wy0 * mx0 * my0);
            s = fmaf(v01, wx1 * wy0 * mx1 * my0, s);
            s = fmaf(v10, wx0 * wy1 * mx0 * my1, s);
            s = fmaf(v11, wx1 * wy1 * mx1 * my1, s);
            acc[p] += s;
        }
        __syncthreads();   // all waves done reading smem[cur] before it is restaged
    }

    // ---- fused mean over views + clip, single coalesced store ----
    float* o = out + ((size_t)(b * J_ + j) * N_ + base_pt);
#pragma unroll
    for (int p = 0; p < PT; ++p) {
        const float m = acc[p] * (1.0f / (float)V_);
        o[p * NTHREADS + tid] = fminf(fmaxf(m, 0.0f), 1.0f);
    }
}

extern "C" void kernel_launch(void* const* d_in, const int* in_sizes, int n_in,
                              void* d_out, int out_size, void* d_ws, size_t ws_size,
                              hipStream_t stream) {
    const float* heatmaps    = (const float*)d_in[0];  // (4,5,15,128,240) f32
    const float* sample_grid = (const float*)d_in[1];  // (4,5,1,128000,2) f32
    float*       out         = (float*)d_out;          // (4,15,128000)    f32

    dim3 grd(B_ * J_ * CH);   // 300 workgroups
    project_sample_kernel<<<grd, NTHREADS, 0, stream>>>(heatmaps, sample_grid, out);
}